// RelationAnchor_19481971655246
// MI455X (gfx1250) — compile-verified
//
#include <hip/hip_runtime.h>

#define NPTS   8192
#define NBATCH 2
#define NCH    128          // feature channels
#define NA     15           // anchors kept
#define TOPK   16           // selected (first is dropped, per idx[:, :, 1:])
#define CHUNK  512          // columns resident in LDS per pass
#define CHUNKP 516          // padded row stride: 16B-aligned rows, lane halves on disjoint banks
#define NCHUNK (NPTS / CHUNK)
#define ROWS   16           // rows per workgroup (one WMMA M-tile)
#define TPB    128          // 4 wave32 waves
#define NWAVE  (TPB / 32)
#define NEG_INF (-__builtin_inff())

typedef __attribute__((ext_vector_type(2))) float v2f;   // 16x4 f32 A/B fragment = 2 VGPRs
typedef __attribute__((ext_vector_type(8))) float v8f;   // 16x16 f32 C/D fragment = 8 VGPRs

// ---------------------------------------------------------------- sq = |l|^2
__global__ void sq_kernel(const float* __restrict__ loc, float* __restrict__ sq) {
  int t = blockIdx.x * blockDim.x + threadIdx.x;
  if (t < NBATCH * NPTS) {
    float x = loc[3 * t], y = loc[3 * t + 1], z = loc[3 * t + 2];
    sq[t] = x * x + y * y + z * z;
  }
}

// wave32 butterfly argmax over (val desc, idx asc, slot asc); all lanes converge
__device__ __forceinline__ void argmax_bfly(float& bv, int& bi, int& bq) {
#pragma unroll
  for (int off = 16; off >= 1; off >>= 1) {
    float ov = __shfl_xor(bv, off, 32);
    int   oi = __shfl_xor(bi, off, 32);
    int   oq = __shfl_xor(bq, off, 32);
    if (ov > bv || (ov == bv && (oi < bi || (oi == bi && oq < bq)))) {
      bv = ov; bi = oi; bq = oq;
    }
  }
}

// ------------------------------------------- fused WMMA distances + top-16
__global__ __launch_bounds__(TPB) void topk_kernel(const float* __restrict__ loc,
                                                   const float* __restrict__ sqg,
                                                   float* __restrict__ locOut,
                                                   int* __restrict__ idxOut) {
  __shared__ float distbuf[ROWS * CHUNKP];  // 33 KB: 16 rows x 512 cols (padded)
  __shared__ float candV[ROWS * 32];        // running 16 + current-chunk 16 per row
  __shared__ int   candI[ROWS * 32];

  const int b    = blockIdx.x >> 9;          // 512 row-tiles per batch
  const int row0 = (blockIdx.x & 511) << 4;
  const int tid  = threadIdx.x;
  const int wave = tid >> 5;
  const int lane = tid & 31;
  const int m    = lane & 15;
  const int kb   = (lane >> 4) << 1;         // K base: 0 (lanes 0-15) or 2 (lanes 16-31)

  const float* Lb  = loc + (size_t)b * NPTS * 3;
  const float* sqb = sqg + (size_t)b * NPTS;

  for (int i = tid; i < ROWS * 32; i += TPB) { candV[i] = NEG_INF; candI[i] = 0x7fffffff; }

  // A fragment (16x4, constant for whole block): [-2x, -2y, -2z, |l|^2]
  // All loads unconditional (valid for every lane) -> pure v_cndmask, no exec branch.
  const int rm = row0 + m;
  v2f afrag;
  afrag.x = -2.0f * Lb[rm * 3 + kb];                        // K = 0 or 2
  {
    const float a1  = -2.0f * Lb[rm * 3 + 1];
    const float asq = sqb[rm];
    afrag.y = (kb == 0) ? a1 : asq;                         // K = 1 or 3 (bias)
  }

  __syncthreads();

  for (int ch = 0; ch < NCHUNK; ++ch) {
    const int cbase = ch * CHUNK;

    // ---- fill LDS distance chunk with V_WMMA_F32_16X16X4_F32 tiles ----
#pragma unroll
    for (int tt = 0; tt < CHUNK / 16 / NWAVE; ++tt) {
      const int t  = wave + tt * NWAVE;
      const int cn = cbase + t * 16 + m;
      v2f bfrag;                                 // B column: [x, y, z, 1]
      bfrag.x = Lb[cn * 3 + kb];
      {
        const float by = Lb[cn * 3 + 1];         // unconditional load, select below
        bfrag.y = (kb == 0) ? by : 1.0f;
      }
      const float sqc = sqb[cn];
      v8f cfrag;
#pragma unroll
      for (int j = 0; j < 8; ++j) cfrag[j] = sqc;  // + |l_n|^2 (column norm) via accumulator
      v8f d = __builtin_amdgcn_wmma_f32_16x16x4_f32(false, afrag, false, bfrag,
                                                    (short)0, cfrag, false, false);
      const int rbase = (lane >> 4) << 3;        // VGPR j holds rows j and j+8
      const int cloc  = t * 16 + m;
#pragma unroll
      for (int j = 0; j < 8; ++j) distbuf[(rbase + j) * CHUNKP + cloc] = d[j];
    }
    __syncthreads();

    // ---- per-row: chunk top-16 (argmax + invalidate), merge into running 16 ----
    for (int r = wave * 4; r < wave * 4 + 4; ++r) {
      float* drow = &distbuf[r * CHUNKP];
      for (int p = 0; p < TOPK; ++p) {
        float bv = NEG_INF; int bi = 0x7fffffff; int bq = 0;
#pragma unroll
        for (int q = lane; q < CHUNK / 4; q += 32) {   // ds_load_b128, 64-bank clean
          const float4 v = *(const float4*)&drow[q * 4];
          const int c = q * 4;
          if (v.x > bv || (v.x == bv && cbase + c     < bi)) { bv = v.x; bi = cbase + c;     bq = c;     }
          if (v.y > bv || (v.y == bv && cbase + c + 1 < bi)) { bv = v.y; bi = cbase + c + 1; bq = c + 1; }
          if (v.z > bv || (v.z == bv && cbase + c + 2 < bi)) { bv = v.z; bi = cbase + c + 2; bq = c + 2; }
          if (v.w > bv || (v.w == bv && cbase + c + 3 < bi)) { bv = v.w; bi = cbase + c + 3; bq = c + 3; }
        }
        argmax_bfly(bv, bi, bq);
        if (lane == 0) {
          drow[bq] = NEG_INF;                    // invalidate winner
          candV[r * 32 + TOPK + p] = bv;
          candI[r * 32 + TOPK + p] = bi;
        }
      }
      // merge 32 candidates (one per lane, register resident) -> sorted 16
      float mv = candV[r * 32 + lane];
      int   mi = candI[r * 32 + lane];
      float selV = NEG_INF; int selI = 0;
      for (int p = 0; p < TOPK; ++p) {
        float bv = mv; int bi = mi; int bq = lane;
        argmax_bfly(bv, bi, bq);
        if (lane == p)  { selV = bv; selI = bi; }
        if (lane == bq) mv = NEG_INF;
      }
      if (lane < TOPK) { candV[r * 32 + lane] = selV; candI[r * 32 + lane] = selI; }
    }
    __syncthreads();
  }

  // ---- epilogue: write indices (drop k=0) and gather anchor locations ----
  for (int r = wave * 4; r < wave * 4 + 4; ++r) {
    if (lane >= 1 && lane < TOPK) {
      const int grow = row0 + r;
      const int sidx = candI[r * 32 + lane];
      const size_t obase = ((size_t)b * NPTS + grow) * NA + (lane - 1);
      idxOut[obase] = sidx;
      locOut[obase * 3 + 0] = Lb[sidx * 3 + 0];
      locOut[obase * 3 + 1] = Lb[sidx * 3 + 1];
      locOut[obase * 3 + 2] = Lb[sidx * 3 + 2];
    }
  }
}

// ------------------------------------------- feature gather (HBM-store bound)
#define FEAT_TOTAL (NBATCH * NCH * NPTS * NA)   // 31,457,280 < 2^31: 32-bit math

__global__ void gather_feat_kernel(const float* __restrict__ feat,
                                   const int* __restrict__ idxOut,
                                   float* __restrict__ featOut) {
  unsigned tid = blockIdx.x * 256u + threadIdx.x;
  if (tid >= (unsigned)FEAT_TOTAL) return;
  unsigned k = tid % 15u;          // magic-multiply
  unsigned t = tid / 15u;
  unsigned n = t & (NPTS - 1);  t >>= 13;
  unsigned c = t & (NCH - 1);
  unsigned b = t >> 7;
  int col = idxOut[((b << 13) + n) * NA + k];
  featOut[tid] = feat[(((b << 7) + c) << 13) + col];   // features L2-resident (8.4 MB)
}

extern "C" void kernel_launch(void* const* d_in, const int* in_sizes, int n_in,
                              void* d_out, int out_size, void* d_ws, size_t ws_size,
                              hipStream_t stream) {
  const float* loc  = (const float*)d_in[0];   // (2, 8192, 3)
  const float* feat = (const float*)d_in[1];   // (2, 128, 8192)

  float* out     = (float*)d_out;
  float* locOut  = out;                                      // 2*8192*15*3 = 737280
  float* featOut = out + (size_t)NBATCH * NPTS * NA * 3;     // 2*128*8192*15
  int*   idxOut  = (int*)(featOut + (size_t)FEAT_TOTAL);     // 2*8192*15 (int32 bits)

  float* sq = (float*)d_ws;                                  // 2*8192 floats = 64 KB

  sq_kernel<<<(NBATCH * NPTS + 255) / 256, 256, 0, stream>>>(loc, sq);
  topk_kernel<<<NBATCH * (NPTS / ROWS), TPB, 0, stream>>>(loc, sq, locOut, idxOut);
  gather_feat_kernel<<<(FEAT_TOTAL + 255) / 256, 256, 0, stream>>>(feat, idxOut, featOut);
}